// TripletLoss_19980187861172
// MI455X (gfx1250) — compile-verified
//
#include <hip/hip_runtime.h>
#include <math.h>

typedef __attribute__((ext_vector_type(2))) float v2f;
typedef __attribute__((ext_vector_type(8))) float v8f;

#define N_ROWS 8192
#define DIMS   512

// ---------------------------------------------------------------------------
// Kernel 1: per-row prep.  sq_x[r] = ||x_r||^2 ; y_n[r,:] = norm_s * y_r/||y_r||
// One 256-thread block per row; each thread owns 2 of the 512 columns.
// ---------------------------------------------------------------------------
__global__ __launch_bounds__(256)
void prep_kernel(const float* __restrict__ x,
                 const float* __restrict__ y,
                 const float* __restrict__ norm_s,
                 float* __restrict__ y_n,
                 float* __restrict__ sq_x) {
    __shared__ float redx[256];
    __shared__ float redy[256];
    const int row = blockIdx.x;
    const int tid = threadIdx.x;
    const float* xr = x + (size_t)row * DIMS;
    const float* yr = y + (size_t)row * DIMS;
    float x0 = xr[tid], x1 = xr[tid + 256];
    float y0 = yr[tid], y1 = yr[tid + 256];
    redx[tid] = x0 * x0 + x1 * x1;
    redy[tid] = y0 * y0 + y1 * y1;
    __syncthreads();
    #pragma unroll
    for (int s = 128; s > 0; s >>= 1) {
        if (tid < s) { redx[tid] += redx[tid + s]; redy[tid] += redy[tid + s]; }
        __syncthreads();
    }
    if (tid == 0) sq_x[row] = redx[0];
    const float scale = norm_s[0] / sqrtf(redy[0]);
    float* ynr = y_n + (size_t)row * DIMS;
    ynr[tid]       = y0 * scale;
    ynr[tid + 256] = y1 * scale;
}

// ---------------------------------------------------------------------------
// Kernel 2: WMMA triplet kernel.  One wave (32 lanes) handles 16 triplets.
// Dot products via V_WMMA_F32_16X16X4_F32; only the diagonal of each 16x16
// tile is consumed:  diag(t) = row_i[t] . row_j[t].
// ---------------------------------------------------------------------------
union V8 { v8f v; float f[8]; };

__device__ __forceinline__ float softplus_stable(float z) {
    // softplus(z) = max(z,0) + log1p(exp(-|z|))
    float az = fabsf(z);
    return (z > 0.0f ? z : 0.0f) + log1pf(expf(-az));
}

__global__ __launch_bounds__(32)
void triplet_wmma_kernel(const float* __restrict__ xmat,   // N x D
                         const float* __restrict__ ymat,   // normalized y, N x D
                         const float* __restrict__ sq_x,   // N
                         const float* __restrict__ norm_s, // 1
                         const int*   __restrict__ trip,   // T x 3
                         int T,
                         float* __restrict__ partials)     // ngroups*2
{
    const int g    = blockIdx.x;
    const int lane = threadIdx.x;          // 0..31, one wave
    const int r    = lane & 15;            // triplet slot within the group
    const int colOff = (lane >> 4) * 2;    // ISA operand layout: upper half-wave
                                           // carries K+2,K+3 of each 16x4 step

    const int  t     = g * 16 + r;
    const bool valid = (t < T);
    const int  tc    = valid ? t : (T - 1);
    const int  ii    = trip[3 * tc + 0];
    const int  jj    = trip[3 * tc + 1];
    const int  kk    = trip[3 * tc + 2];

    // Which lanes hold a diagonal element of the 16x16 C/D tile, and where.
    const bool active = (lane < 8) || (lane >= 24);
    const int  comp   = (lane < 8) ? lane : (lane - 24);

    __shared__ float red[32];

    const float ns  = norm_s[0];
    const float ns2 = ns * ns;
    float sums[2];

    #pragma unroll
    for (int phase = 0; phase < 2; ++phase) {
        const float* mat = (phase == 0) ? xmat : ymat;
        // Per-lane gathered row base pointers (A: i-rows; B: j-rows / k-rows).
        const float* pa = mat + (size_t)ii * DIMS + colOff;
        const float* pj = mat + (size_t)jj * DIMS + colOff;
        const float* pk = mat + (size_t)kk * DIMS + colOff;

        v8f cj = {}; v8f ck = {};
        #pragma unroll 8
        for (int k = 0; k < DIMS; k += 4) {
            v2f a  = *(const v2f*)(pa + k);   // A 16x4 operand (2 VGPRs)
            v2f bj = *(const v2f*)(pj + k);   // B 4x16 operand (2 VGPRs)
            v2f bk = *(const v2f*)(pk + k);
            cj = __builtin_amdgcn_wmma_f32_16x16x4_f32(
                    false, a, false, bj, (short)0, cj, false, false);
            ck = __builtin_amdgcn_wmma_f32_16x16x4_f32(
                    false, a, false, bk, (short)0, ck, false, false);
        }

        // Extract this lane's diagonal entries and evaluate the triplet term.
        V8 uj; uj.v = cj;
        V8 uk; uk.v = ck;
        float dj = uj.f[comp & 7];
        float dk = uk.f[comp & 7];

        float contrib = 0.0f;
        if (active && valid) {
            float sqi, sqj, sqk;
            if (phase == 0) { sqi = sq_x[ii]; sqj = sq_x[jj]; sqk = sq_x[kk]; }
            else            { sqi = ns2;      sqj = ns2;      sqk = ns2;      }
            float dij = fmaxf(sqi + sqj - 2.0f * dj, 0.0f);
            float dik = fmaxf(sqi + sqk - 2.0f * dk, 0.0f);
            contrib = softplus_stable(dij - dik);
        }
        red[lane] = contrib;
        __syncthreads();
        float s = 0.0f;
        if (lane == 0) {
            #pragma unroll
            for (int q = 0; q < 32; ++q) s += red[q];
        }
        sums[phase] = s;
        __syncthreads();
    }

    if (lane == 0) {
        partials[2 * g + 0] = sums[0];   // image-loss partial
        partials[2 * g + 1] = sums[1];   // text-loss partial
    }
}

// ---------------------------------------------------------------------------
// Kernel 3: deterministic fixed-order reduction of per-group partials.
// ---------------------------------------------------------------------------
__global__ __launch_bounds__(256)
void finalize_kernel(const float* __restrict__ partials, int ngroups,
                     float invT, float* __restrict__ out) {
    __shared__ float simg[256];
    __shared__ float stxt[256];
    const int tid = threadIdx.x;
    float a = 0.0f, b = 0.0f;
    for (int g = tid; g < ngroups; g += 256) {
        a += partials[2 * g + 0];
        b += partials[2 * g + 1];
    }
    simg[tid] = a; stxt[tid] = b;
    __syncthreads();
    #pragma unroll
    for (int s = 128; s > 0; s >>= 1) {
        if (tid < s) { simg[tid] += simg[tid + s]; stxt[tid] += stxt[tid + s]; }
        __syncthreads();
    }
    if (tid == 0) out[0] = (simg[0] + stxt[0]) * invT;
}

// ---------------------------------------------------------------------------
extern "C" void kernel_launch(void* const* d_in, const int* in_sizes, int n_in,
                              void* d_out, int out_size, void* d_ws, size_t ws_size,
                              hipStream_t stream) {
    const float* x      = (const float*)d_in[0];
    const float* y      = (const float*)d_in[1];
    const float* norm_s = (const float*)d_in[2];
    const int*   trip   = (const int*)d_in[3];
    const int    T      = in_sizes[3] / 3;

    // Workspace layout: y_n (N*D f32) | sq_x (N f32) | partials (ngroups*2 f32)
    char*  ws       = (char*)d_ws;
    float* y_n      = (float*)ws;
    float* sq_x     = (float*)(ws + (size_t)N_ROWS * DIMS * sizeof(float));
    float* partials = (float*)(ws + (size_t)N_ROWS * DIMS * sizeof(float)
                                  + (size_t)N_ROWS * sizeof(float));

    const int ngroups = (T + 15) / 16;

    prep_kernel<<<N_ROWS, 256, 0, stream>>>(x, y, norm_s, y_n, sq_x);
    triplet_wmma_kernel<<<ngroups, 32, 0, stream>>>(x, y_n, sq_x, norm_s, trip,
                                                    T, partials);
    finalize_kernel<<<1, 256, 0, stream>>>(partials, ngroups,
                                           1.0f / (float)T, (float*)d_out);
}